// RoiMultiHead_25469156065605
// MI455X (gfx1250) — compile-verified
//
#include <hip/hip_runtime.h>

// Problem constants (from reference)
#define BN      32768   // batch
#define DD      2048    // feature dim
#define NROI    16
#define RR      8       // reg outputs
#define HH      2
#define CC      10
#define NCOLS   28      // R + H*C
#define WP      48      // padded n-width of pair-interleaved weight layout
#define PRS     (2 * WP)            // floats per pair-row (96) -> +1 pair-row = 32 banks
#define KCHUNK  128     // K tile through LDS
#define NCHUNK  (DD / KCHUNK)
#define LSTR    132     // padded LDS row stride for A (floats)
#define ABUFB   (16 * LSTR * 4)          // 8448 bytes per A buffer
#define BBUFB   ((KCHUNK / 2) * PRS * 4) // 24576 bytes per B buffer
#define MAXTILES 2064   // ceil(B/16) + 16 ragged slack

typedef __attribute__((ext_vector_type(2))) float v2f;
typedef __attribute__((ext_vector_type(8))) float v8f;

// ---- CDNA5 async memory->LDS copy (ASYNCcnt path) ----
__device__ __forceinline__ void async_b128(unsigned lds_byte_addr, const float* gaddr) {
    asm volatile("global_load_async_to_lds_b128 %0, %1, off"
                 :: "v"(lds_byte_addr), "v"(gaddr) : "memory");
}
__device__ __forceinline__ void wait_async0() {
    asm volatile("s_wait_asynccnt 0x0" ::: "memory");
}

// ---------------- bucketing ----------------

__global__ void roi_init_kernel(int* __restrict__ counts) {
    if (threadIdx.x < 32) counts[threadIdx.x] = 0;   // counts[16] + cursors[16]
}

__global__ void roi_count_kernel(const int* __restrict__ roi_ids,
                                 int* __restrict__ counts) {
    int b = blockIdx.x * blockDim.x + threadIdx.x;
    if (b < BN) atomicAdd(&counts[roi_ids[b] & (NROI - 1)], 1);
}

__global__ void roi_prefix_kernel(const int* __restrict__ counts,
                                  int* __restrict__ offsets) {
    if (threadIdx.x == 0) {
        int a = 0;
        for (int e = 0; e < NROI; ++e) { offsets[e] = a; a += counts[e]; }
        offsets[NROI] = a;
    }
}

__global__ void roi_scatter_kernel(const int* __restrict__ roi_ids,
                                   const int* __restrict__ offsets,
                                   int* __restrict__ cursors,
                                   int* __restrict__ perm) {
    int b = blockIdx.x * blockDim.x + threadIdx.x;
    if (b < BN) {
        int e = roi_ids[b] & (NROI - 1);
        int p = offsets[e] + atomicAdd(&cursors[e], 1);
        perm[p] = b;
    }
}

// ---- weight packing: pair-interleaved [NROI][DD/2][WP][2] ----
// element (k, n) of the 2048x32 panel sits at (k>>1)*PRS + 2n + (k&1);
// cols 0..7 = reg head, 8..27 = cls heads (h*10+c), 28..47 = zero pad.

__global__ void pack_w_kernel(const float* __restrict__ Wreg,
                              const float* __restrict__ Wcls,
                              float* __restrict__ packW) {
    int idx = blockIdx.x * blockDim.x + threadIdx.x;   // < NROI*(DD/2)*PRS
    int e   = idx / ((DD / 2) * PRS);
    int rem = idx - e * ((DD / 2) * PRS);
    int r2  = rem / PRS;                 // pair-row (k>>1)
    int t   = rem - r2 * PRS;
    int n   = t >> 1;
    int d   = r2 * 2 + (t & 1);          // k
    float v = 0.0f;
    if (n < RR) {
        v = Wreg[((size_t)e * DD + d) * RR + n];
    } else if (n < NCOLS) {
        int hc = n - RR;
        int h  = hc / CC;
        int cc = hc - h * CC;
        v = Wcls[(((size_t)e * HH + h) * DD + d) * CC + cc];
    }
    packW[idx] = v;
}

// ---------------- main WMMA GEMM over roi buckets ----------------

__global__ __launch_bounds__(64) void roi_gemm_kernel(
        const float* __restrict__ feats,
        const float* __restrict__ breg,
        const float* __restrict__ bcls,
        const int*   __restrict__ offsets,
        const int*   __restrict__ perm,
        const float* __restrict__ packW,
        float*       __restrict__ out) {
    __shared__ __align__(16) float lA[2][16 * LSTR];            // A tile: [m][k], padded
    __shared__ __align__(16) float lB[2][(KCHUNK / 2) * PRS];   // B tile: pair-interleaved
    __shared__ int sRow[16];
    __shared__ int sMeta[3];                                    // roi, base, nvalid

    const int tid = threadIdx.x;

    if (tid == 0) {
        int t = blockIdx.x, acc = 0, roi = -1, tib = 0;
        for (int e = 0; e < NROI; ++e) {
            int cnt   = offsets[e + 1] - offsets[e];
            int tiles = (cnt + 15) >> 4;
            if (t < acc + tiles) { roi = e; tib = t - acc; break; }
            acc += tiles;
        }
        int base = 0, nv = 0;
        if (roi >= 0) {
            base = offsets[roi] + tib * 16;
            nv   = offsets[roi + 1] - base;
            if (nv > 16) nv = 16;
        }
        sMeta[0] = roi; sMeta[1] = base; sMeta[2] = nv;
    }
    __syncthreads();
    const int roi = sMeta[0];
    if (roi < 0) return;                        // block-uniform exit
    const int base = sMeta[1];
    const int nv   = sMeta[2];
    if (tid < 16) sRow[tid] = perm[base + (tid < nv ? tid : nv - 1)];
    __syncthreads();

    const int lane = tid & 31;
    const int wv   = tid >> 5;                  // wave 0 -> cols 0..15, wave 1 -> 16..31
    const int mRow = lane & 15;
    const int hi   = lane >> 4;                 // lanes 16..31 take K+2,K+3
    const int ksel = hi << 1;
    const int nIdx = wv * 16 + (lane & 15);

    // ---- per-thread async copy slots ----
    // A: 8 x b128 per thread; row r = (tid>>5) + 2j, float4 col = tid&31
    const int  r0 = tid >> 5;
    const int  c4 = tid & 31;
    const float* gA[8];
#pragma unroll
    for (int j = 0; j < 8; ++j)
        gA[j] = feats + (size_t)sRow[r0 + 2 * j] * DD + 4 * c4;
    const unsigned lA0 = (unsigned)(uintptr_t)&lA[0][0] + (unsigned)((r0 * LSTR + 4 * c4) * 4);
    // B: 24 x b128 per thread; chunk is one contiguous 24KB region of packW
    const float* wbase = packW + (size_t)roi * (DD / 2) * PRS;
    const float* gB0   = wbase + tid * 4;
    const unsigned lB0 = (unsigned)(uintptr_t)&lB[0][0] + (unsigned)(tid * 16);

    v8f acc = {};                               // C/D: VGPR j -> M=j (lanes<16) / j+8

    // prologue: chunk 0 -> buffer 0
#pragma unroll
    for (int j = 0; j < 8; ++j)  async_b128(lA0 + 1056u * j, gA[j]);
#pragma unroll
    for (int j = 0; j < 24; ++j) async_b128(lB0 + 1024u * j, gB0 + 256 * j);
    wait_async0();
    __syncthreads();

    for (int c = 0; c < NCHUNK; ++c) {
        const int cur = c & 1;
        const int nxt = cur ^ 1;
        if (c + 1 < NCHUNK) {                   // issue next chunk (no wait yet)
            const int k0n = (c + 1) * KCHUNK;
#pragma unroll
            for (int j = 0; j < 8; ++j)
                async_b128(lA0 + (unsigned)(nxt * ABUFB) + 1056u * j, gA[j] + k0n);
#pragma unroll
            for (int j = 0; j < 24; ++j)
                async_b128(lB0 + (unsigned)(nxt * BBUFB) + 1024u * j,
                           gB0 + (size_t)(k0n / 2) * PRS + 256 * j);
        }
        // compute current chunk
        const float* lAc = &lA[cur][mRow * LSTR + ksel];
        const float* lBc = &lB[cur][(hi * PRS) + 2 * nIdx];
#pragma unroll
        for (int kk = 0; kk < KCHUNK; kk += 4) {
            v2f a = *(const v2f*)(lAc + kk);
            v2f b = *(const v2f*)(lBc + (kk >> 1) * PRS);
            acc = __builtin_amdgcn_wmma_f32_16x16x4_f32(
                    false, a, false, b, (short)0, acc, false, false);
        }
        wait_async0();                           // next-chunk copies landed
        __syncthreads();                         // everyone done with cur + nxt ready
    }

    // store: lane holds D[m][n] with m = j + 8*(lane>=16), n = nIdx
    const int mhi = hi << 3;
    if (nIdx < NCOLS) {
        const float bias = (nIdx < RR) ? breg[roi * RR + nIdx]
                                       : bcls[roi * (HH * CC) + (nIdx - RR)];
#pragma unroll
        for (int j = 0; j < 8; ++j) {
            int m = j + mhi;
            if (m < nv) {
                int s = sRow[m];
                float val = acc[j] + bias;
                if (nIdx < RR) out[(size_t)s * RR + nIdx] = val;
                else           out[(size_t)BN * RR + (size_t)s * (HH * CC) + (nIdx - RR)] = val;
            }
        }
    }
}

// ---------------- launch ----------------

extern "C" void kernel_launch(void* const* d_in, const int* in_sizes, int n_in,
                              void* d_out, int out_size, void* d_ws, size_t ws_size,
                              hipStream_t stream) {
    const float* feats   = (const float*)d_in[0];
    const float* Wreg    = (const float*)d_in[1];
    const float* breg    = (const float*)d_in[2];
    const float* Wcls    = (const float*)d_in[3];
    const float* bcls    = (const float*)d_in[4];
    const int*   roi_ids = (const int*)d_in[5];
    float*       out     = (float*)d_out;

    // workspace: counts[16] | cursors[16] | offsets[17..pad] | perm[BN] | packW
    int* counts  = (int*)d_ws;
    int* cursors = counts + 16;
    int* offsets = counts + 32;
    int* perm    = counts + 64;
    float* packW = (float*)(counts + 64 + BN);   // byte offset 131328, 16B aligned

    roi_init_kernel   <<<1, 64, 0, stream>>>(counts);
    pack_w_kernel     <<<(NROI * (DD / 2) * PRS) / 256, 256, 0, stream>>>(Wreg, Wcls, packW);
    roi_count_kernel  <<<BN / 256, 256, 0, stream>>>(roi_ids, counts);
    roi_prefix_kernel <<<1, 32, 0, stream>>>(counts, offsets);
    roi_scatter_kernel<<<BN / 256, 256, 0, stream>>>(roi_ids, offsets, cursors, perm);
    roi_gemm_kernel   <<<MAXTILES, 64, 0, stream>>>(feats, breg, bcls,
                                                    offsets, perm, packW, out);
}